// CompressedSensingInception_33131377722033
// MI455X (gfx1250) — compile-verified
//
#include <hip/hip_runtime.h>
#include <math.h>

typedef float v2f __attribute__((ext_vector_type(2)));
typedef float v8f __attribute__((ext_vector_type(8)));

#define ALPHA 0.3f
#define BN_SCALE 0.9995003749f  /* 1/sqrt(1.001) */

__device__ __forceinline__ float leaky(float x) { return x >= 0.f ? x : ALPHA * x; }

// D = A(16x4) * B(4x16) + C, f32 WMMA (CDNA5 dedicated fp32 matrix pipe)
static __device__ __forceinline__ v8f wmma4(v2f a, v2f b, v8f c) {
#if __has_builtin(__builtin_amdgcn_wmma_f32_16x16x4_f32)
  return __builtin_amdgcn_wmma_f32_16x16x4_f32(false, a, false, b, (short)0, c, false, false);
#else
  c[0] += a.x * b.x;
  c[1] += a.y * b.y;
  return c;
#endif
}

// ---------------------------------------------------------------------------
// workspace layout (floats)
#define WS_Y      0
#define WS_YLAST  497664
#define WS_IM     995328
#define WS_LAM    1003104
#define WS_T1     1003200           /* post-conv scratch; doubles as P1/P2 */
#define WS_T2     1666752
#define WS_T3     1998528
#define OUT1      106272            /* 32*9*9*41; cs_out starts here in d_out */
// P1: GEMM1 B operand, padded+pair-interleaved: [2592][96][2]  = 497664 f
// P2: GEMM2 B operand (mat^T), padded+pair-interleaved: [42][5184][2] = 435456 f
// Both live in the t1..t3 region (only needed after FISTA finishes).

// ---------------------------------------------------------------------------
__global__ void zero_kernel(float* __restrict__ p, int n) {
  int i = blockIdx.x * 256 + threadIdx.x;
  if (i < n) p[i] = 0.f;
}

// Pack mat into guard-free, coalesced-by-lane layouts (runs once per call).
__global__ void pack_kernel(const float* __restrict__ mat,
                            float* __restrict__ p1, float* __restrict__ p2) {
  int idx = blockIdx.x * 256 + threadIdx.x;
  if (idx < 5184 * 96) {  // p1[(k>>1)*192 + col*2 + (k&1)] = mat[k][col]
    int k = idx / 96, col = idx % 96;
    float v = (col < 81) ? mat[(size_t)k * 81 + col] : 0.f;
    p1[(size_t)(k >> 1) * 192 + col * 2 + (k & 1)] = v;
  }
  if (idx < 84 * 5184) {  // p2[(k>>1)*10368 + n*2 + (k&1)] = mat[n][k]
    int k = idx / 5184, n = idx % 5184;
    float v = (k < 81) ? mat[(size_t)n * 81 + k] : 0.f;
    p2[(size_t)(k >> 1) * 10368 + n * 2 + (k & 1)] = v;
  }
}

// ---------------------------------------------------------------------------
struct PrepArgs {
  const float *inp, *w1_k, *w1_b, *x1_k, *x1_b;
  const float *y17_k, *y17_b, *y71_k, *y71_b, *yc_k, *yc_b;
  const float *d1_k, *d2_k, *h1_w, *h1_b, *h2_w, *h2_b, *h3_w, *h3_b;
  float *im, *lam, *out;
};

// One block per batch element, 96 threads. Computes w-branch (out ch0),
// y-branch (out ch9..40), im (B*3 x 81) and lam (B*3) for FISTA.
__global__ __launch_bounds__(96) void prep_kernel(PrepArgs A) {
  __shared__ float s_inp[243];
  __shared__ float s_xn[243];
  __shared__ float s_max[27];
  __shared__ float s_z1[108];
  __shared__ float s_z2[24];
  __shared__ float s_a1[24];
  __shared__ float s_a2[12];
  __shared__ float s_y1[81 * 32];
  __shared__ float s_y2[81 * 32];

  const int tid = threadIdx.x;
  const int b = blockIdx.x;
  const float* in_b = A.inp + b * 243;
  for (int e = tid; e < 243; e += 96) s_inp[e] = in_b[e];
  __syncthreads();

  // ---- w branch: 1x1 conv 3->1 + leaky -> out channel 0
  if (tid < 81) {
    int i = tid / 9, j = tid % 9;
    float acc = A.w1_b[0];
    for (int c = 0; c < 3; ++c) acc += s_inp[(i * 9 + j) * 3 + c] * A.w1_k[c];
    A.out[((b * 9 + i) * 9 + j) * 41 + 0] = leaky(acc);
  }

  // ---- x normalize: max over axis 1 (height) per (j,c)
  if (tid < 27) {
    int j = tid / 3, c = tid % 3;
    float m = 0.f;
    for (int i = 0; i < 9; ++i) m = fmaxf(m, fabsf(s_inp[(i * 9 + j) * 3 + c]));
    s_max[tid] = m;
  }
  __syncthreads();
  for (int e = tid; e < 243; e += 96) {
    int j = (e / 3) % 9, c = e % 3;
    s_xn[e] = s_inp[e] / (0.001f + s_max[j * 3 + c]);
  }
  __syncthreads();
  // x1: 1x1 conv 3->3 + leaky; store im[(b*3+c)*81 + n]
  for (int e = tid; e < 243; e += 96) {
    int c = e / 81, n = e % 81;
    float acc = A.x1_b[c];
    for (int ci = 0; ci < 3; ++ci) acc += s_xn[n * 3 + ci] * A.x1_k[ci * 3 + c];
    A.im[(b * 3 + c) * 81 + n] = leaky(acc);
  }

  // ---- z branch: d1 5x5 s3 (9->3), d2 5x5 s3 (3->1), MLP 24->24->12->1
  if (tid < 108) {
    int oy = tid / 36, ox = (tid / 12) % 3, oc = tid % 12;
    float acc = 0.f;
    for (int ky = 0; ky < 5; ++ky) {
      int iy = oy * 3 - 1 + ky;
      if (iy < 0 || iy >= 9) continue;
      for (int kx = 0; kx < 5; ++kx) {
        int ix = ox * 3 - 1 + kx;
        if (ix < 0 || ix >= 9) continue;
        for (int ci = 0; ci < 3; ++ci)
          acc += s_inp[(iy * 9 + ix) * 3 + ci] * BN_SCALE *
                 A.d1_k[((ky * 5 + kx) * 3 + ci) * 12 + oc];
      }
    }
    s_z1[tid] = leaky(acc * BN_SCALE);
  }
  __syncthreads();
  if (tid < 24) {
    float acc = 0.f;
    for (int ky = 0; ky < 5; ++ky) {
      int iy = -1 + ky;
      if (iy < 0 || iy >= 3) continue;
      for (int kx = 0; kx < 5; ++kx) {
        int ix = -1 + kx;
        if (ix < 0 || ix >= 3) continue;
        for (int ci = 0; ci < 12; ++ci)
          acc += s_z1[(iy * 3 + ix) * 12 + ci] * A.d2_k[((ky * 5 + kx) * 12 + ci) * 24 + tid];
      }
    }
    s_z2[tid] = leaky(acc * BN_SCALE);
  }
  __syncthreads();
  if (tid < 24) {
    float acc = A.h1_b[tid];
    for (int i = 0; i < 24; ++i) acc += s_z2[i] * A.h1_w[i * 24 + tid];
    s_a1[tid] = acc;
  }
  __syncthreads();
  if (tid < 12) {
    float acc = A.h2_b[tid];
    for (int i = 0; i < 24; ++i) acc += s_a1[i] * A.h2_w[i * 12 + tid];
    s_a2[tid] = acc;
  }
  __syncthreads();
  if (tid == 0) {
    float acc = A.h3_b[0];
    for (int i = 0; i < 12; ++i) acc += s_a2[i] * A.h3_w[i];
    float param = 0.1f / (1.f + expf(-acc));
    float lamv = param * 0.1f;
    A.lam[b * 3 + 0] = lamv;
    A.lam[b * 3 + 1] = lamv;
    A.lam[b * 3 + 2] = lamv;
  }

  // ---- y branch: 1x7 and 7x1 convs (3->32 each), concat, 1x1 64->32, leaky
  for (int e = tid; e < 2592; e += 96) {
    int o = e % 32, pos = e / 32;
    int i = pos / 9, j = pos % 9;
    float a1v = A.y17_b[o], a2v = A.y71_b[o];
    for (int k = 0; k < 7; ++k) {
      int jj = j + k - 3, ii = i + k - 3;
      for (int c = 0; c < 3; ++c) {
        if (jj >= 0 && jj < 9) a1v += s_inp[(i * 9 + jj) * 3 + c] * A.y17_k[(k * 3 + c) * 32 + o];
        if (ii >= 0 && ii < 9) a2v += s_inp[(ii * 9 + j) * 3 + c] * A.y71_k[(k * 3 + c) * 32 + o];
      }
    }
    s_y1[pos * 32 + o] = a1v;
    s_y2[pos * 32 + o] = a2v;
  }
  __syncthreads();
  for (int e = tid; e < 2592; e += 96) {
    int o = e % 32, pos = e / 32;
    int i = pos / 9, j = pos % 9;
    float acc = A.yc_b[o];
    for (int ci = 0; ci < 32; ++ci) acc += s_y1[pos * 32 + ci] * A.yc_k[ci * 32 + o];
    for (int ci = 0; ci < 32; ++ci) acc += s_y2[pos * 32 + ci] * A.yc_k[(32 + ci) * 32 + o];
    A.out[((b * 9 + i) * 9 + j) * 41 + 9 + o] = leaky(acc);
  }
}

// ---------------------------------------------------------------------------
// Persistent FISTA: 6 workgroups x 8 waves, 100 iterations, workgroup-local
// sync only. Guard-free b64 operand loads from packed p1/p2; dual WMMA
// accumulator chains in GEMM2 for XDL ILP.
__global__ __launch_bounds__(256) void fista_kernel(const float* __restrict__ p1,
                                                    const float* __restrict__ p2,
                                                    const float* __restrict__ im,
                                                    const float* __restrict__ lam_g,
                                                    float* __restrict__ y,
                                                    float* __restrict__ ylast,
                                                    float* __restrict__ out) {
  __shared__ float red[8][6][256];   // per-wave partial C tiles (GEMM1)
  __shared__ float resid[16][96];    // 16x81 residual, zero-padded to 96
  __shared__ float im_s[16][96];     // im tile cached (constant across iters)
  __shared__ float lam_s[16];

  const int tid = threadIdx.x;
  const int lane = tid & 31;
  const int wave = __builtin_amdgcn_readfirstlane(threadIdx.x >> 5);  // SGPR
  const int r0 = blockIdx.x * 16;    // first global row of this tile
  const int mrow = lane & 15;
  const int khalf = lane >> 4;       // frag K group: lanes 16-31 hold K+2
  const int colb = lane & 15;

  if (tid < 16) lam_s[tid] = lam_g[r0 + tid];
  for (int e = tid; e < 1536; e += 256) {
    int m = e / 96, n = e % 96;
    im_s[m][n] = (n < 81) ? im[(size_t)(r0 + m) * 81 + n] : 0.f;
  }
  __syncthreads();

  float t = 1.0f;
#pragma unroll 1
  for (int it = 0; it < 100; ++it) {
    // ---------------- GEMM1 partials: K slice [wave*648, wave*648+648) -----
    v8f acc[6] = {};
    const int kbeg = wave * 648, kend = kbeg + 648;  // scalar bounds
#pragma unroll 1
    for (int kk = kbeg; kk < kend; kk += 4) {
      if ((kk & 127) == 0)  // scalar branch: stage upcoming p1 K-rows
        __builtin_prefetch(p1 + (size_t)((kk + 256) >> 1) * 192, 0, 3);
      const int k0 = kk + khalf * 2;                  // even
      v2f a = *(const v2f*)(y + (size_t)(r0 + mrow) * 5184 + k0);
      const float* bp = p1 + (size_t)(k0 >> 1) * 192 + colb * 2;
#pragma unroll
      for (int j = 0; j < 6; ++j) {
        v2f bf = *(const v2f*)(bp + j * 32);          // single b64, coalesced
        acc[j] = wmma4(a, bf, acc[j]);
      }
    }
    // spill partials to LDS
#pragma unroll
    for (int j = 0; j < 6; ++j)
#pragma unroll
      for (int r = 0; r < 8; ++r)
        red[wave][j][(r + khalf * 8) * 16 + colb] = acc[j][r];
    __syncthreads();

    // reduce 8 wave partials: resid = im - y@mat (cols 81..95 stay zero)
    for (int e = tid; e < 1536; e += 256) {
      int m = e / 96, n = e % 96;
      float s = 0.f;
#pragma unroll
      for (int w = 0; w < 8; ++w) s += red[w][n >> 4][m * 16 + (n & 15)];
      resid[m][n] = im_s[m][n] - s;
    }
    __syncthreads();

    // ---------------- GEMM2 + fused FISTA update --------------------------
    float tn = (1.f + sqrtf(1.f + 4.f * t * t)) * 0.5f;
    float coef = (t - 1.f) / tn;
    t = tn;
#pragma unroll 1
    for (int j2 = wave; j2 < 324; j2 += 8) {  // scalar loop
      v8f ca = {}, cb = {};                   // two independent WMMA chains
      const int ncol = j2 * 16 + colb;
      const float* bp2 = p2 + (size_t)ncol * 2;
#pragma unroll
      for (int s = 0; s < 21; ++s) {
        const int k0 = s * 4 + khalf * 2;     // even
        v2f a = *(const v2f*)(&resid[mrow][k0]);           // ds_load_b64
        v2f bf = *(const v2f*)(bp2 + (size_t)(k0 >> 1) * 10368);  // b64
        if (s & 1) cb = wmma4(a, bf, cb);
        else       ca = wmma4(a, bf, ca);
      }
      v8f c2 = ca + cb;
      // epilogue: wv = y + re; soft-threshold; momentum; write y/y_last
#pragma unroll
      for (int r = 0; r < 8; ++r) {
        int m = r + khalf * 8;
        size_t idx = (size_t)(r0 + m) * 5184 + ncol;
        float lamv = lam_s[m];
        float wv = y[idx] + c2[r];
        float yn = fmaxf(wv - lamv, 0.f) - fmaxf(-wv - lamv, 0.f);
        float yl = ylast[idx];
        ylast[idx] = yn;
        y[idx] = yn + coef * (yn - yl);
      }
    }
    __threadfence_block();
    __syncthreads();
  }

  // cs_out: (B,72,72,3) from y_new (== ylast after final iteration)
  for (int e = tid; e < 16 * 5184; e += 256) {
    int m = e / 5184, n = e % 5184;
    int row = r0 + m, b = row / 3, c = row - b * 3;
    out[OUT1 + ((size_t)b * 5184 + n) * 3 + c] = ylast[(size_t)row * 5184 + n];
  }
}

// ---------------------------------------------------------------------------
// Post-CS conv chain (tiny: ~0.35 GFLOP total, plain VALU)
__global__ void c51_kernel(const float* __restrict__ cs, const float* __restrict__ k,
                           const float* __restrict__ bias, float* __restrict__ t1) {
  int idx = blockIdx.x * 256 + threadIdx.x;
  if (idx >= 32 * 36 * 36 * 16) return;
  int oc = idx & 15, tt = idx >> 4;
  int ow = tt % 36; tt /= 36;
  int oh = tt % 36; int b = tt / 36;
  float acc = bias[oc];
  for (int kh = 0; kh < 5; ++kh) {
    int ih = oh * 2 - 1 + kh;
    if (ih < 0 || ih >= 72) continue;
    const float* p = cs + ((size_t)(b * 72 + ih) * 72 + ow * 2) * 3;
    for (int ci = 0; ci < 3; ++ci) acc += p[ci] * k[(kh * 3 + ci) * 16 + oc];
  }
  t1[idx] = acc;
}

__global__ void c15_kernel(const float* __restrict__ t1, const float* __restrict__ k,
                           const float* __restrict__ bias, float* __restrict__ t2) {
  int idx = blockIdx.x * 256 + threadIdx.x;
  if (idx >= 32 * 18 * 18 * 32) return;
  int oc = idx & 31, tt = idx >> 5;
  int ow = tt % 18; tt /= 18;
  int oh = tt % 18; int b = tt / 18;
  float acc = bias[oc];
  for (int kw = 0; kw < 5; ++kw) {
    int iw = ow * 2 - 1 + kw;
    if (iw < 0 || iw >= 36) continue;
    const float* p = t1 + ((size_t)(b * 36 + oh * 2) * 36 + iw) * 16;
    for (int ci = 0; ci < 16; ++ci) acc += p[ci] * k[(kw * 16 + ci) * 32 + oc];
  }
  t2[idx] = acc;
}

__global__ void c55_kernel(const float* __restrict__ t2, const float* __restrict__ k,
                           const float* __restrict__ bias, float* __restrict__ t3) {
  int idx = blockIdx.x * 256 + threadIdx.x;
  if (idx >= 32 * 9 * 9 * 64) return;
  int oc = idx & 63, tt = idx >> 6;
  int ow = tt % 9; tt /= 9;
  int oh = tt % 9; int b = tt / 9;
  float acc = bias[oc];
  for (int kh = 0; kh < 5; ++kh) {
    int ih = oh * 2 - 1 + kh;
    if (ih < 0 || ih >= 18) continue;
    for (int kw = 0; kw < 5; ++kw) {
      int iw = ow * 2 - 1 + kw;
      if (iw < 0 || iw >= 18) continue;
      const float* p = t2 + ((size_t)(b * 18 + ih) * 18 + iw) * 32;
      for (int ci = 0; ci < 32; ++ci) acc += p[ci] * k[((kh * 5 + kw) * 32 + ci) * 64 + oc];
    }
  }
  t3[idx] = acc;
}

__global__ void x2_kernel(const float* __restrict__ t3, const float* __restrict__ k,
                          const float* __restrict__ bias, float* __restrict__ out) {
  int idx = blockIdx.x * 256 + threadIdx.x;
  if (idx >= 32 * 81 * 8) return;
  int oc = idx & 7, tt = idx >> 3;
  int pos = tt % 81, b = tt / 81;
  float acc = bias[oc];
  const float* p = t3 + (size_t)(b * 81 + pos) * 64;
  for (int ci = 0; ci < 64; ++ci) acc += p[ci] * k[ci * 8 + oc];
  out[(size_t)(b * 81 + pos) * 41 + 1 + oc] = leaky(acc);
}

// ---------------------------------------------------------------------------
extern "C" void kernel_launch(void* const* d_in, const int* in_sizes, int n_in,
                              void* d_out, int out_size, void* d_ws, size_t ws_size,
                              hipStream_t stream) {
  const float* inp   = (const float*)d_in[0];
  const float* mat   = (const float*)d_in[1];
  const float* w1_k  = (const float*)d_in[2];
  const float* w1_b  = (const float*)d_in[3];
  const float* x1_k  = (const float*)d_in[4];
  const float* x1_b  = (const float*)d_in[5];
  const float* c51_k = (const float*)d_in[6];
  const float* c51_b = (const float*)d_in[7];
  const float* c15_k = (const float*)d_in[8];
  const float* c15_b = (const float*)d_in[9];
  const float* c55_k = (const float*)d_in[10];
  const float* c55_b = (const float*)d_in[11];
  const float* x2_k  = (const float*)d_in[12];
  const float* x2_b  = (const float*)d_in[13];
  const float* y17_k = (const float*)d_in[14];
  const float* y17_b = (const float*)d_in[15];
  const float* y71_k = (const float*)d_in[16];
  const float* y71_b = (const float*)d_in[17];
  const float* yc_k  = (const float*)d_in[18];
  const float* yc_b  = (const float*)d_in[19];
  const float* d1_k  = (const float*)d_in[20];
  const float* d2_k  = (const float*)d_in[21];
  const float* h1_w  = (const float*)d_in[22];
  const float* h1_b  = (const float*)d_in[23];
  const float* h2_w  = (const float*)d_in[24];
  const float* h2_b  = (const float*)d_in[25];
  const float* h3_w  = (const float*)d_in[26];
  const float* h3_b  = (const float*)d_in[27];

  float* out = (float*)d_out;
  float* ws  = (float*)d_ws;
  float* y     = ws + WS_Y;
  float* ylast = ws + WS_YLAST;
  float* im    = ws + WS_IM;
  float* lam   = ws + WS_LAM;
  float* t1    = ws + WS_T1;
  float* t2    = ws + WS_T2;
  float* t3    = ws + WS_T3;
  float* p1    = ws + WS_T1;            // overlaps conv scratch (safe: pack ->
  float* p2    = ws + WS_T1 + 497664;   // fista -> convs are stream-ordered)

  // zero FISTA state (ws is poisoned by the harness)
  zero_kernel<<<(995328 + 255) / 256, 256, 0, stream>>>(ws, 995328);
  // pack mat into guard-free WMMA operand layouts
  pack_kernel<<<(5184 * 96 + 255) / 256, 256, 0, stream>>>(mat, p1, p2);

  PrepArgs pa;
  pa.inp = inp; pa.w1_k = w1_k; pa.w1_b = w1_b; pa.x1_k = x1_k; pa.x1_b = x1_b;
  pa.y17_k = y17_k; pa.y17_b = y17_b; pa.y71_k = y71_k; pa.y71_b = y71_b;
  pa.yc_k = yc_k; pa.yc_b = yc_b; pa.d1_k = d1_k; pa.d2_k = d2_k;
  pa.h1_w = h1_w; pa.h1_b = h1_b; pa.h2_w = h2_w; pa.h2_b = h2_b;
  pa.h3_w = h3_w; pa.h3_b = h3_b;
  pa.im = im; pa.lam = lam; pa.out = out;
  prep_kernel<<<32, 96, 0, stream>>>(pa);

  // persistent WMMA-f32 FISTA: 6 row-tiles x 8 waves, 100 local iterations
  fista_kernel<<<6, 256, 0, stream>>>(p1, p2, im, lam, y, ylast, out);

  const float* cs = out + OUT1;  // (32,72,72,3) written by fista_kernel
  c51_kernel<<<(32 * 36 * 36 * 16 + 255) / 256, 256, 0, stream>>>(cs, c51_k, c51_b, t1);
  c15_kernel<<<(32 * 18 * 18 * 32 + 255) / 256, 256, 0, stream>>>(t1, c15_k, c15_b, t2);
  c55_kernel<<<(32 * 9 * 9 * 64 + 255) / 256, 256, 0, stream>>>(t2, c55_k, c55_b, t3);
  x2_kernel<<<(32 * 81 * 8 + 255) / 256, 256, 0, stream>>>(t3, x2_k, x2_b, out);
}